// GCNModel_44032004718866
// MI455X (gfx1250) — compile-verified
//
#include <hip/hip_runtime.h>
#include <math.h>

#define NN 100000
#define NE 1600000
#define ET (NE + NN)   /* edges incl. self loops = 1,700,000 */

typedef __attribute__((ext_vector_type(2))) float v2f;
typedef __attribute__((ext_vector_type(8))) float v8f;

__device__ __forceinline__ void atomic_max_f32(float* addr, float v) {
  // IEEE trick: signed max for v>=0, unsigned min for v<0. Works with -inf init.
  if (v >= 0.0f) atomicMax((int*)addr, __float_as_int(v));
  else           atomicMin((unsigned int*)addr, (unsigned int)__float_as_int(v));
}

__device__ __forceinline__ void atomic_add_f32(float* addr, float v) {
  unsafeAtomicAdd(addr, v);   // native global_atomic_add_f32
}

// ---------------- GEMM via V_WMMA_F32_16X16X4_F32 ----------------
// One wave computes one 16x16 tile of H = X[N x Fin] @ W[Fin x Fout].
__global__ void k_gemm_wmma(const float* __restrict__ X, const float* __restrict__ W,
                            float* __restrict__ H, int Fin, int Fout, int rowTiles) {
  int wave = (int)((blockIdx.x * blockDim.x + threadIdx.x) >> 5);
  int colTiles = Fout >> 4;
  int totalTiles = rowTiles * colTiles;
  if (wave >= totalTiles) return;            // wave-uniform -> EXEC stays all-ones
  int rt = wave / colTiles;
  int ct = wave - rt * colTiles;
  int row0 = rt << 4, col0 = ct << 4;
  int lane = threadIdx.x & 31;
  int half = lane >> 4;                       // 0: K lo pair, 1: K hi pair
  int r    = lane & 15;
  v8f c = {};
  const float* xrow = X + (size_t)(row0 + r) * Fin;
  for (int k0 = 0; k0 < Fin; k0 += 4) {
    int ka = k0 + 2 * half;
    v2f a, b;
    a.x = xrow[ka];
    a.y = xrow[ka + 1];
    b.x = W[(size_t)ka * Fout + col0 + r];
    b.y = W[(size_t)(ka + 1) * Fout + col0 + r];
    c = __builtin_amdgcn_wmma_f32_16x16x4_f32(false, a, false, b, (short)0, c,
                                              false, false);
  }
#pragma unroll
  for (int v = 0; v < 8; ++v)
    H[(size_t)(row0 + v + 8 * half) * Fout + col0 + r] = c[v];
}

// Per-node attention logits: ssrc[i] = h_i . a_src ; sdst[i] = h_i . a_dst
__global__ void k_logits(const float* __restrict__ H, const float* __restrict__ asrc,
                         const float* __restrict__ adst, float* __restrict__ ssrc,
                         float* __restrict__ sdst, int F) {
  int node = (int)((blockIdx.x * blockDim.x + threadIdx.x) >> 5);
  if (node >= NN) return;
  int lane = threadIdx.x & 31;
  float vs = 0.f, vd = 0.f;
  for (int f = lane; f < F; f += 32) {
    float h = H[(size_t)node * F + f];
    vs += h * asrc[f];
    vd += h * adst[f];
  }
#pragma unroll
  for (int off = 16; off > 0; off >>= 1) {
    vs += __shfl_down(vs, off);
    vd += __shfl_down(vd, off);
  }
  if (lane == 0) { ssrc[node] = vs; sdst[node] = vd; }
}

__global__ void k_fill(float* __restrict__ p, int n, float v) {
  int i = blockIdx.x * blockDim.x + threadIdx.x;
  if (i < n) p[i] = v;
}

__device__ __forceinline__ void edge_sd(const int* __restrict__ ei, int e, int& s, int& d) {
  if (e < NE) { s = ei[e]; d = ei[NE + e]; }
  else        { s = d = e - NE; }
}

__global__ void k_edge_max(const int* __restrict__ ei, const float* __restrict__ ssrc,
                           const float* __restrict__ sdst, float* __restrict__ m) {
  int e = blockIdx.x * blockDim.x + threadIdx.x;
  if (e >= ET) return;
  int s, d; edge_sd(ei, e, s, d);
  float t = ssrc[s] + sdst[d];
  t = t > 0.f ? t : 0.2f * t;               // leaky_relu, slope 0.2
  atomic_max_f32(&m[d], t);
}

__global__ void k_edge_expsum(const int* __restrict__ ei, const float* __restrict__ ssrc,
                              const float* __restrict__ sdst, const float* __restrict__ m,
                              float* __restrict__ ex, float* __restrict__ denom) {
  int e = blockIdx.x * blockDim.x + threadIdx.x;
  if (e >= ET) return;
  int s, d; edge_sd(ei, e, s, d);
  float t = ssrc[s] + sdst[d];
  t = t > 0.f ? t : 0.2f * t;
  float v = expf(t - m[d]);
  ex[e] = v;
  atomic_add_f32(&denom[d], v);
}

// One lane per (edge, feature): coalesced gather of h[src] row, coalesced atomics on accum[dst].
__global__ void k_edge_scatter(const int* __restrict__ ei, const float* __restrict__ ex,
                               const float* __restrict__ denom, const float* __restrict__ H,
                               float* __restrict__ accum, int F, int logF) {
  long long tid = (long long)blockIdx.x * blockDim.x + threadIdx.x;
  long long total = (long long)ET << logF;
  if (tid >= total) return;
  int e = (int)(tid >> logF);
  int f = (int)(tid & (F - 1));
  int s, d; edge_sd(ei, e, s, d);
  float alpha = ex[e] / denom[d];
  atomic_add_f32(&accum[(size_t)d * F + f], alpha * H[(size_t)s * F + f]);
}

__global__ void k_bias_relu(float* __restrict__ a, const float* __restrict__ b,
                            int F, int total) {
  int i = blockIdx.x * blockDim.x + threadIdx.x;
  if (i >= total) return;
  float v = a[i] + b[i & (F - 1)];
  a[i] = v > 0.f ? v : 0.f;
}

// Column sums of accum[N x 32] -> osum[32] (lane == feature).
__global__ void k_colsum(const float* __restrict__ accum, float* __restrict__ osum) {
  __shared__ float tmp[8][32];
  int lane = threadIdx.x & 31;
  int wid  = threadIdx.x >> 5;
  float s = 0.f;
  int stride = gridDim.x * 8;
  for (int row = blockIdx.x * 8 + wid; row < NN; row += stride)
    s += accum[(size_t)row * 32 + lane];
  tmp[wid][lane] = s;
  __syncthreads();
  if (wid == 0) {
    float t = 0.f;
#pragma unroll
    for (int w = 0; w < 8; ++w) t += tmp[w][lane];
    atomic_add_f32(&osum[lane], t);
  }
}

__global__ void k_final(const float* __restrict__ osum, const float* __restrict__ b3,
                        const float* __restrict__ thr, const float* __restrict__ can,
                        float* __restrict__ out) {
  int f = threadIdx.x;
  if (f < 32) {
    float td = thr[0] * can[0];
    out[f] = td * (osum[f] / (float)NN + b3[f]);
  }
}

extern "C" void kernel_launch(void* const* d_in, const int* in_sizes, int n_in,
                              void* d_out, int out_size, void* d_ws, size_t ws_size,
                              hipStream_t stream) {
  (void)in_sizes; (void)n_in; (void)out_size; (void)ws_size;
  const float* x   = (const float*)d_in[0];
  const int*   ei  = (const int*)d_in[1];
  const float* W1  = (const float*)d_in[2];
  const float* as1 = (const float*)d_in[3];
  const float* ad1 = (const float*)d_in[4];
  const float* b1  = (const float*)d_in[5];
  const float* W2  = (const float*)d_in[6];
  const float* as2 = (const float*)d_in[7];
  const float* ad2 = (const float*)d_in[8];
  const float* b2  = (const float*)d_in[9];
  const float* W3  = (const float*)d_in[10];
  const float* as3 = (const float*)d_in[11];
  const float* ad3 = (const float*)d_in[12];
  const float* b3  = (const float*)d_in[13];
  const float* thr = (const float*)d_in[14];
  const float* can = (const float*)d_in[15];
  float* out = (float*)d_out;

  char* ws = (char*)d_ws;
  size_t off = 0;
  auto carve = [&](size_t nfloats) {
    float* p = (float*)(ws + off);
    off += ((nfloats * 4 + 255) / 256) * 256;
    return p;
  };
  float* bufA = carve((size_t)NN * 64);  // transformed features h = X@W
  float* bufB = carve((size_t)NN * 64);  // aggregation accumulator / next layer input
  float* ssrc = carve(NN);
  float* sdst = carve(NN);
  float* mbuf = carve(NN);
  float* dbuf = carve(NN);
  float* exb  = carve(ET);
  float* osum = carve(32);

  auto run_layer = [&](const float* Xin, int Fin, int Fout,
                       const float* W, const float* asv, const float* adv) {
    int rowTiles = NN / 16;                          // 6250 (exact)
    int totalTiles = rowTiles * (Fout / 16);
    k_gemm_wmma<<<(totalTiles + 7) / 8, 256, 0, stream>>>(Xin, W, bufA, Fin, Fout, rowTiles);
    k_logits<<<(NN + 7) / 8, 256, 0, stream>>>(bufA, asv, adv, ssrc, sdst, Fout);
    k_fill<<<(NN + 255) / 256, 256, 0, stream>>>(mbuf, NN, -__builtin_inff());
    k_fill<<<(NN + 255) / 256, 256, 0, stream>>>(dbuf, NN, 0.f);
    int nacc = NN * Fout;
    k_fill<<<(nacc + 255) / 256, 256, 0, stream>>>(bufB, nacc, 0.f);
    k_edge_max<<<(ET + 255) / 256, 256, 0, stream>>>(ei, ssrc, sdst, mbuf);
    k_edge_expsum<<<(ET + 255) / 256, 256, 0, stream>>>(ei, ssrc, sdst, mbuf, exb, dbuf);
    long long total = (long long)ET * Fout;
    int logF = (Fout == 64) ? 6 : 5;
    k_edge_scatter<<<(int)((total + 255) / 256), 256, 0, stream>>>(ei, exb, dbuf, bufA,
                                                                   bufB, Fout, logF);
  };

  // Layer 1: 64 -> 64, relu
  run_layer(x, 64, 64, W1, as1, ad1);
  k_bias_relu<<<(NN * 64 + 255) / 256, 256, 0, stream>>>(bufB, b1, 64, NN * 64);
  // Layer 2: 64 -> 32, relu
  run_layer(bufB, 64, 32, W2, as2, ad2);
  k_bias_relu<<<(NN * 32 + 255) / 256, 256, 0, stream>>>(bufB, b2, 32, NN * 32);
  // Layer 3: 32 -> 32, scale by td, mean over nodes
  run_layer(bufB, 32, 32, W3, as3, ad3);
  k_fill<<<1, 32, 0, stream>>>(osum, 32, 0.f);
  k_colsum<<<512, 256, 0, stream>>>(bufB, osum);
  k_final<<<1, 32, 0, stream>>>(osum, b3, thr, can, out);
}